// GAPNet_Module_26173530702024
// MI455X (gfx1250) — compile-verified
//
#include <hip/hip_runtime.h>
#include <hip/hip_bf16.h>

// ---------------------------------------------------------------------------
// GAPNet module for MI455X (gfx1250, wave32).
//   B=8, N=4096, C=3, F=128, NUM_POINTS=32, RADIUS=0.6
// Kernel 1: ball query via wave32 ballot scan (ordered first-32 selection,
//           uniform early exit).
// Kernel 2: per-point fused pipeline; edge features via v_wmma_f32_16x16x4_f32
//           (K=C=3 padded to 4), bias fused into the WMMA accumulator,
//           ef tile staged in LDS with conflict-free 129-dword row stride.
// ---------------------------------------------------------------------------

#define BATCH 8
#define NPTS 4096
#define NFEAT 128
#define KNBR 32
#define R2 0.36f
#define EF_STRIDE 129  // 128 + 1 dword pad -> bank = (row + col) % 64, conflict-free

typedef float v2f __attribute__((ext_vector_type(2)));
typedef float v8f __attribute__((ext_vector_type(8)));

// ---------------------------------------------------------------------------
// Kernel 1: one wave per query point. Scan candidates in index order, 32 per
// iteration (lane = candidate). Ballot + prefix-popcount assigns output slots
// in increasing-index order, exactly matching sort(where(d>R2, N, idx))[:32].
// Pad with the first valid index (matches gi[:, :, :1] fill).
// ---------------------------------------------------------------------------
__global__ __launch_bounds__(256) void ball_query_kernel(
    const float* __restrict__ x, int* __restrict__ gi) {
  const int lane = threadIdx.x & 31;
  const int wid = threadIdx.x >> 5;
  const int p = blockIdx.x * 8 + wid;          // global point id
  const int b = p >> 12;                        // / 4096
  const int n = p & (NPTS - 1);

  const float* xb = x + (size_t)b * NPTS * 3;
  const float xq0 = xb[n * 3 + 0];
  const float xq1 = xb[n * 3 + 1];
  const float xq2 = xb[n * 3 + 2];
  const float sqn = xq0 * xq0 + xq1 * xq1 + xq2 * xq2;

  int* go = gi + (size_t)p * KNBR;
  int cnt = 0;        // wave-uniform
  int firstIdx = -1;  // wave-uniform

  for (int chunk = 0; chunk < (NPTS / 32) && cnt < KNBR; ++chunk) {
    const int m = chunk * 32 + lane;
    const float m0 = xb[m * 3 + 0];
    const float m1 = xb[m * 3 + 1];
    const float m2 = xb[m * 3 + 2];
    const float sqm = m0 * m0 + m1 * m1 + m2 * m2;
    const float dot = xq0 * m0 + xq1 * m1 + xq2 * m2;
    const float d = sqn + sqm - 2.0f * dot;     // same formula as reference
    const bool pred = !(d > R2);
    const unsigned mask = __builtin_amdgcn_ballot_w32(pred);
    if (firstIdx < 0 && mask != 0u)
      firstIdx = chunk * 32 + __builtin_ctz(mask);
    const int before = __builtin_popcount(mask & ((1u << lane) - 1u));
    const int pos = cnt + before;
    if (pred && pos < KNBR) go[pos] = m;
    cnt += __builtin_popcount(mask);
  }
  // fill remaining slots with first valid index (self is always valid)
  if (cnt < KNBR && lane >= cnt) go[lane] = firstIdx;
}

__device__ __forceinline__ float wave_sum(float v) {
  for (int off = 16; off > 0; off >>= 1) v += __shfl_xor(v, off, 32);
  return v;
}
__device__ __forceinline__ float wave_max(float v) {
  for (int off = 16; off > 0; off >>= 1) v = fmaxf(v, __shfl_xor(v, off, 32));
  return v;
}

// ---------------------------------------------------------------------------
// Kernel 2: one wave per point; 2 waves (2 points) per 64-thread block.
// ---------------------------------------------------------------------------
__global__ __launch_bounds__(64) void gapnet_main_kernel(
    const float* __restrict__ x, const int* __restrict__ gi,
    const float* __restrict__ W1, const float* __restrict__ b1,
    const float* __restrict__ W2, const float* __restrict__ b2,
    const float* __restrict__ W3, const float* __restrict__ b3,
    const float* __restrict__ W4, const float* __restrict__ b4,
    float* __restrict__ out) {
  __shared__ float ef[2][KNBR * EF_STRIDE];  // per-wave 32x128 ef tile (padded)
  __shared__ float coefs[2][KNBR];

  const int lane = threadIdx.x & 31;
  const int wid = threadIdx.x >> 5;
  const int p = blockIdx.x * 2 + wid;
  const int b = p >> 12;
  const int n = p & (NPTS - 1);

  const float* xb = x + (size_t)b * NPTS * 3;
  const float xq0 = xb[n * 3 + 0];
  const float xq1 = xb[n * 3 + 1];
  const float xq2 = xb[n * 3 + 2];

  // ---- self attention: relu(W3 . relu(W1 x + b1) + b3), fused per wave ----
  float sa_acc = 0.0f;
  #pragma unroll
  for (int j = 0; j < 4; ++j) {
    const int f = lane + 32 * j;
    float nf = W1[f * 3 + 0] * xq0 + W1[f * 3 + 1] * xq1 +
               W1[f * 3 + 2] * xq2 + b1[f];
    nf = fmaxf(nf, 0.0f);
    sa_acc += nf * W3[f];
  }
  const float self_att = fmaxf(wave_sum(sa_acc) + b3[0], 0.0f);

  // ---- edge vector: lane = neighbor k ----
  const int idx = gi[(size_t)p * KNBR + lane];
  const float e0 = xq0 - xb[idx * 3 + 0];
  const float e1 = xq1 - xb[idx * 3 + 1];
  const float e2 = xq2 - xb[idx * 3 + 2];

  // ---- edge features via WMMA f32 16x16x4: ef = relu(edge(32x4) @ W2^T(4x128) + b2)
  // A 16x4 layout: lanes 0-15 hold M rows {VGPR0=K0, VGPR1=K1};
  //                lanes 16-31 hold      {VGPR0=K2, VGPR1=K3(=0 pad)}.
  // B 4x16 layout: VGPR0: lanes0-15 K0 row, lanes16-31 K2 row;
  //                VGPR1: lanes0-15 K1 row, lanes16-31 K3 row(=0).
  const int halfLane = lane & 15;
  const bool loHalf = lane < 16;
  #pragma unroll
  for (int mt = 0; mt < 2; ++mt) {
    const int srcLane = mt * 16 + halfLane;
    const float g0 = __shfl(e0, srcLane, 32);
    const float g1 = __shfl(e1, srcLane, 32);
    const float g2 = __shfl(e2, srcLane, 32);
    v2f a;
    a.x = loHalf ? g0 : g2;
    a.y = loHalf ? g1 : 0.0f;
    #pragma unroll
    for (int t = 0; t < 8; ++t) {
      const int fb = t * 16;
      const int fn = fb + halfLane;
      v2f bm;
      bm.x = loHalf ? W2[fn * 3 + 0] : W2[fn * 3 + 2];
      bm.y = loHalf ? W2[fn * 3 + 1] : 0.0f;
      const float cv = b2[fn];  // bias fused into accumulator (same column)
      v8f c = {cv, cv, cv, cv, cv, cv, cv, cv};
      v8f d = __builtin_amdgcn_wmma_f32_16x16x4_f32(
          false, a, false, bm, (short)0, c, false, false);
      const int rowBase = mt * 16 + (loHalf ? 0 : 8);
      #pragma unroll
      for (int r = 0; r < 8; ++r) {
        const float v = fmaxf(d[r], 0.0f);
        ef[wid][(rowBase + r) * EF_STRIDE + fn] = v;
      }
    }
  }
  __syncthreads();  // ef tile visible across lanes

  // ---- neighbor attention + softmax over k (lane = k) ----
  const float* myrow = &ef[wid][lane * EF_STRIDE];
  float att = 0.0f;
  #pragma unroll 8
  for (int f = 0; f < NFEAT; ++f) att += myrow[f] * W4[f];
  const float na = fmaxf(att + b4[0], 0.0f);
  const float logit = self_att + na;
  const float lr = (logit > 0.0f) ? logit : 0.01f * logit;  // leaky_relu(0.01)
  const float mx = wave_max(lr);
  const float ex = __expf(lr - mx);
  const float sm = wave_sum(ex);
  coefs[wid][lane] = ex / sm;
  __syncthreads();

  // ---- vals[f] = sum_k coef[k]*ef[k][f]; elu; store out[b][f][n] ----
  // 16MB output fits entirely in the 192MB L2, so the stride-4096 column
  // stores coalesce into full-line HBM writebacks from L2.
  #pragma unroll
  for (int j = 0; j < 4; ++j) {
    const int f = lane + 32 * j;
    float v = 0.0f;
    #pragma unroll 8
    for (int k = 0; k < KNBR; ++k)
      v += coefs[wid][k] * ef[wid][k * EF_STRIDE + f];
    const float o = (v > 0.0f) ? v : (__expf(v) - 1.0f);  // elu
    out[((size_t)b * NFEAT + f) * NPTS + n] = o;
  }
}

extern "C" void kernel_launch(void* const* d_in, const int* in_sizes, int n_in,
                              void* d_out, int out_size, void* d_ws,
                              size_t ws_size, hipStream_t stream) {
  (void)in_sizes; (void)n_in; (void)out_size; (void)ws_size;
  const float* x = (const float*)d_in[0];
  const float* W1 = (const float*)d_in[1];
  const float* b1 = (const float*)d_in[2];
  const float* W2 = (const float*)d_in[3];
  const float* b2 = (const float*)d_in[4];
  const float* W3 = (const float*)d_in[5];
  const float* b3 = (const float*)d_in[6];
  const float* W4 = (const float*)d_in[7];
  const float* b4 = (const float*)d_in[8];
  float* out = (float*)d_out;
  int* gi = (int*)d_ws;  // B*N*32 ints = 4 MB

  const int total_points = BATCH * NPTS;  // 32768
  ball_query_kernel<<<total_points / 8, 256, 0, stream>>>(x, gi);
  gapnet_main_kernel<<<total_points / 2, 64, 0, stream>>>(
      x, gi, W1, b1, W2, b2, W3, b3, W4, b4, out);
}